// MNN_18219251270051
// MI455X (gfx1250) — compile-verified
//
#include <hip/hip_runtime.h>
#include <cstdint>
#include <cstddef>

#ifndef __has_builtin
#define __has_builtin(x) 0
#endif

#if __has_builtin(__builtin_amdgcn_global_load_async_to_lds_b32)
#define HAVE_ASYNC_LDS 1
#else
#define HAVE_ASYNC_LDS 0
#endif

#if __has_builtin(__builtin_amdgcn_s_wait_asynccnt)
#define HAVE_WAIT_ASYNC 1
#else
#define HAVE_WAIT_ASYNC 0
#endif

// Problem constants (from reference: B=8, Cin=8, H=128, O=32, K=5)
#define BQ    8
#define CIN   8
#define HH    128
#define OO    32
#define KK    5
#define FH    124          // H - K + 1
#define TI    4            // output rows per block
#define TROWS (TI + KK - 1)  // 8 input rows staged
#define JS    31           // output cols per block (124 = 4*31)

// Builtin signature (from hipcc diagnostic): first arg is AS(1) int*, so
// LDS arg is AS(3) int*.
typedef __attribute__((address_space(1))) int* gas_iptr;
typedef __attribute__((address_space(3))) int* lds_iptr;

__global__ __launch_bounds__(256)
void minmax_conv_kernel(const float* __restrict__ x,
                        const float* __restrict__ Kh,
                        const float* __restrict__ Km,
                        float* __restrict__ out)
{
    // x tile: [row 0..7][col 0..127][c 0..7], 8192 floats = 32 KB
    __shared__ float xt[TROWS * HH * CIN];

    const int tid  = threadIdx.x;          // 0..255
    const int bb   = blockIdx.y;           // batch 0..7
    const int tile = blockIdx.x >> 2;      // 0..30
    const int jq   = blockIdx.x & 3;       // 0..3
    const int i0   = tile * TI;            // first output row (<=120)
    const int j0   = jq * JS;              // first output col (0,31,62,93)

    // ---------------- stage x[b, :, i0..i0+7, :] into LDS ----------------
    {
        const float* xb = x + ((size_t)bb * CIN * HH * HH);
#pragma unroll
        for (int k = 0; k < 32; ++k) {
            int idx = (k << 8) | tid;      // 0..8191, unique per (k,tid)
            int c   = idx >> 10;           // 0..7
            int rc  = idx & 1023;
            int row = rc >> 7;             // 0..7
            int col = rc & 127;            // 0..127
            const float* gp = xb + (((size_t)c * HH + (size_t)(i0 + row)) << 7) + col;
            float*       lp = &xt[(((row << 7) + col) << 3) + c];
#if HAVE_ASYNC_LDS
            // CDNA5 async copy: per-lane global addr -> per-lane LDS addr,
            // tracked by ASYNCcnt (no VGPR data return).
            __builtin_amdgcn_global_load_async_to_lds_b32(
                (gas_iptr)(uintptr_t)gp, (lds_iptr)lp, 0, 0);
#else
            *lp = *gp;
#endif
        }
    }
#if HAVE_ASYNC_LDS
#if HAVE_WAIT_ASYNC
    __builtin_amdgcn_s_wait_asynccnt(0);
#else
    asm volatile("s_wait_asynccnt 0" ::: "memory");
#endif
#endif
    __syncthreads();

    // ---------------- per-thread (o, c) pair; taps in registers ----------------
    const int o = tid >> 3;   // 0..31
    const int c = tid & 7;    // 0..7

    float kh[KK * KK], km[KK * KK];
    {
        const float* hb = Kh + (size_t)(o * CIN + c) * (KK * KK);
        const float* mb = Km + (size_t)(o * CIN + c) * (KK * KK);
#pragma unroll
        for (int t = 0; t < KK * KK; ++t) {
            kh[t] = hb[t];
            km[t] = mb[t];
        }
    }

    // out flat layout: (((b*O + o)*FH + i)*FH + j)*CIN + c
    float* ob = out + (((size_t)bb * OO + o) * FH) * FH * CIN;

    for (int j = j0; j < j0 + JS; ++j) {
        // 8x5 column strip of the staged tile (covers 4 output rows)
        float v[TROWS][KK];
#pragma unroll
        for (int rr = 0; rr < TROWS; ++rr) {
#pragma unroll
            for (int q = 0; q < KK; ++q) {
                v[rr][q] = xt[((((rr) << 7) + (j + q)) << 3) + c];
            }
        }

#pragma unroll
        for (int r = 0; r < TI; ++r) {
            float mh, mm;
#pragma unroll
            for (int t = 0; t < KK * KK; ++t) {
                const float w  = v[r + t / KK][t % KK];
                const float dh = w - kh[t];
                const float dm = w - km[t];
                if (t == 0) { mh = dh; mm = dm; }
                else        { mh = fminf(mh, dh); mm = fmaxf(mm, dm); }
            }
            const size_t oi = ((size_t)(i0 + r) * FH + (size_t)j) * CIN + (size_t)c;
            __builtin_nontemporal_store(mh - mm, ob + oi);
        }
    }
}

extern "C" void kernel_launch(void* const* d_in, const int* in_sizes, int n_in,
                              void* d_out, int out_size, void* d_ws, size_t ws_size,
                              hipStream_t stream)
{
    const float* x  = (const float*)d_in[0];   // (8, 8, 128, 128) f32
    const float* Kh = (const float*)d_in[1];   // (32, 8, 5, 5) f32
    const float* Km = (const float*)d_in[2];   // (32, 8, 5, 5) f32
    float*       out = (float*)d_out;          // 31,490,048 f32

    dim3 grid(124, 8);   // x: 31 row-tiles * 4 j-quarters, y: batch
    minmax_conv_kernel<<<grid, 256, 0, stream>>>(x, Kh, Km, out);
}